// CLTBernoulliDecoder_71579924955480
// MI455X (gfx1250) — compile-verified
//
#include <hip/hip_runtime.h>
#include <math.h>

typedef __attribute__((ext_vector_type(2))) float v2f;
typedef __attribute__((ext_vector_type(8))) float v8f;

#define NF      784          // features (tree length)
#define JK      1568         // 2*NF, columns of W
#define ZD      64           // z dim
#define BXM     256          // rows of x -> M of output
#define BZN     4096         // rows of z -> N of output
#define NCHUNK  13           // ceil(784 / 64) j-chunks
#define CHUNK_K 256          // fused K rows per chunk: 128 (a*ls) + 128 (c*lsn)
#define KTOT    (NCHUNK * CHUNK_K)   // 3328 (zero-padded K)
#define NT      16           // i-tile width per workgroup

// ---------------------------------------------------------------------------
// Pass 1: build A workspace in K-PAIR-interleaved layout:
//   A2p[pair][m][kl]  with  pair = t/2, kl = t&1,   flat: pair*(BXM*2)+m*2+kl
// so a WMMA A-fragment (V0 = K even, V1 = K odd, same column m) is ONE
// aligned float2 per lane -> global_load_b64, 16 lanes = 128 contiguous B.
//   t = chunk*256 + g*128 + (jj*2 + k),  j = chunk*64 + jj
//   g==0 : a[b,j,k] = x_cond[b,j,k] *  x[b,j]
//   g==1 : c[b,j,k] = x_cond[b,j,k] * (1 - x[b,j])
//   x_cond[b,j,0] = 1 - x[b, tree[j]] ; x_cond[b,j,1] = x[b, tree[j]]
// Padded rows (j >= 784) are zero so the B side there is don't-care.
// ---------------------------------------------------------------------------
__global__ __launch_bounds__(BXM)
void clt_build_a_kernel(const float* __restrict__ x,
                        const int*   __restrict__ tree,
                        float*       __restrict__ A2p) {
    const int t = blockIdx.x;
    const int b = threadIdx.x;
    const int r  = t & 255;
    const int g  = r >> 7;
    const int jk = r & 127;
    const int k  = jk & 1;
    const int jj = jk >> 1;
    const int j  = (t >> 8) * 64 + jj;
    float val = 0.0f;
    if (j < NF) {
        int jt = tree[j];
        if (jt < 0) jt += NF;                 // x[:, -1] -> last column
        const float xt = x[b * NF + jt];
        const float xc = k ? xt : (1.0f - xt);
        const float xb = x[b * NF + j];
        val = (g == 0) ? (xc * xb) : (xc * (1.0f - xb));
    }
    A2p[(size_t)(t >> 1) * (BXM * 2) + b * 2 + (t & 1)] = val;
}

// ---------------------------------------------------------------------------
// Pass 2: fused  logits -> log_sigmoid -> fp32 WMMA GEMM
// One workgroup per 16-wide i-tile, full M=256. 256 threads = 8 waves (wave32).
// ---------------------------------------------------------------------------
__global__ __launch_bounds__(256)
void clt_fused_kernel(const float* __restrict__ z,
                      const float* __restrict__ W,
                      const float* __restrict__ bias,
                      const float* __restrict__ A2p,
                      float*       __restrict__ out) {
    __shared__ float z_lds[NT][ZD];               // 4 KB: z tile (16 x 64)
    __shared__ float Braw[128][NT];               // 8 KB: raw logits
    __shared__ float Bt[CHUNK_K / 2][NT][2];      // 16 KB: K-pair B-tile

    const int tid  = threadIdx.x;
    const int i0   = blockIdx.x * NT;             // first i (out column)
    const int lane = tid & 31;
    const int wave = tid >> 5;
    const int lm   = lane & 15;
    const int half = lane >> 4;

    // stage z tile: 16 rows x 64 cols, coalesced
    #pragma unroll
    for (int q = 0; q < 4; ++q) {
        const int idx = tid * 4 + q;              // 0..1023
        const int row = idx >> 6, col = idx & 63;
        z_lds[row][col] = z[(size_t)(i0 + row) * ZD + col];
    }
    __syncthreads();

    v8f c0 = {};                                  // wave's two 16x16 C tiles
    v8f c1 = {};
    const int m0 = wave * 32;

    const int il  = tid >> 4;                     // i index, logit phase
    const int l16 = tid & 15;                     // jk sub-lane, logit phase

    for (int chunk = 0; chunk < NCHUNK; ++chunk) {
        // ---- phase 1: raw logits for (16 i) x (128 jk), 8 dots per thread
        const int jkbase = chunk * 128;
        float acc[8];
        #pragma unroll
        for (int q = 0; q < 8; ++q) acc[q] = 0.0f;
        #pragma unroll 4
        for (int d = 0; d < ZD; ++d) {
            const float zz = z_lds[il][d];
            const float* wrow = W + (size_t)d * JK + jkbase + l16;
            #pragma unroll
            for (int q = 0; q < 8; ++q) {
                const int col = jkbase + q * 16 + l16;
                const float wv = (col < JK) ? wrow[q * 16] : 0.0f;
                acc[q] = fmaf(zz, wv, acc[q]);
            }
        }
        #pragma unroll
        for (int q = 0; q < 8; ++q) {
            const int col = jkbase + q * 16 + l16;
            const float bv = (col < JK) ? bias[col] : 0.0f;
            Braw[q * 16 + l16][il] = acc[q] + bv;
        }
        __syncthreads();

        // ---- phase 2: root substitution + log_sigmoid -> Bt = [ls ; lsn]
        #pragma unroll
        for (int q = 0; q < 8; ++q) {
            const int idx = tid + q * 256;        // 0..2047
            const int jk  = idx >> 4;             // ls row (0..127)
            const int i   = idx & 15;
            const int k   = jk & 1;
            const int jj  = jk >> 1;
            const int jg  = chunk * 64 + jj;
            float l;
            if (k == 0) l = (jg == 0) ? Braw[jk + 1][i] : Braw[jk][i];
            else        l = Braw[jk][i];
            if (jg >= NF) l = 0.0f;               // padded region (A is zero)
            const float ax = fabsf(l);
            const float sp = log1pf(__expf(-ax)); // softplus(-|l|)
            // ls at fused row jk -> pair jk/2 ; lsn at row 128+jk -> pair 64+jk/2
            Bt[jk >> 1][i][k]        = fminf(l, 0.0f) - sp;   // log_sigmoid(l)
            Bt[64 + (jk >> 1)][i][k] = fminf(-l, 0.0f) - sp;  // log_sigmoid(-l)
        }
        __syncthreads();

        // ---- phase 3: fp32 WMMA K-loop (K = 256 per chunk, 4 per wmma)
        // Fragment: V0 = K even, V1 = K odd -> one float2 per lane.
        const v2f* Abase = (const v2f*)A2p + (size_t)chunk * (CHUNK_K / 2) * BXM;
        for (int k0 = 0; k0 < CHUNK_K; k0 += 4) {
            const int pr = (k0 >> 1) + half;      // K-pair row for this lane half
            const v2f bf = *(const v2f*)&Bt[pr][lm][0];        // ds_load_b64
            const v2f* ap = Abase + (size_t)pr * BXM + lm;
            const v2f a0 = ap[m0];                              // global b64
            const v2f a1 = ap[m0 + 16];
            c0 = __builtin_amdgcn_wmma_f32_16x16x4_f32(
                    false, a0, false, bf, (short)0, c0, false, false);
            c1 = __builtin_amdgcn_wmma_f32_16x16x4_f32(
                    false, a1, false, bf, (short)0, c1, false, false);
        }
        __syncthreads();
    }

    // ---- epilogue: C layout -> out[m, i0 + n], out is (256 x 4096) row-major
    #pragma unroll
    for (int r = 0; r < 8; ++r) {
        const int mA = m0 + r + 8 * half;
        const int mB = m0 + 16 + r + 8 * half;
        out[(size_t)mA * BZN + i0 + lm] = c0[r];
        out[(size_t)mB * BZN + i0 + lm] = c1[r];
    }
}

// ---------------------------------------------------------------------------
extern "C" void kernel_launch(void* const* d_in, const int* in_sizes, int n_in,
                              void* d_out, int out_size, void* d_ws, size_t ws_size,
                              hipStream_t stream) {
    const float* x    = (const float*)d_in[0];   // (256, 784)
    const float* z    = (const float*)d_in[1];   // (4096, 64)
    const float* W    = (const float*)d_in[2];   // (64, 1568)
    const float* bias = (const float*)d_in[3];   // (1568,)
    const int*   tree = (const int*)d_in[4];     // (784,)
    float* out = (float*)d_out;                  // (256, 4096)

    float* A2p = (float*)d_ws;                   // KTOT x 256 fp32 (~3.3 MB)

    clt_build_a_kernel<<<KTOT, BXM, 0, stream>>>(x, tree, A2p);
    clt_fused_kernel<<<BZN / NT, 256, 0, stream>>>(z, W, bias, A2p, out);
}